// FusionLoss_33182917329290
// MI455X (gfx1250) — compile-verified
//
#include <hip/hip_runtime.h>
#include <hip/hip_bf16.h>

typedef float v2f __attribute__((ext_vector_type(2)));
typedef float v8f __attribute__((ext_vector_type(8)));

static constexpr int IMG_H = 512;
static constexpr int IMG_W = 512;
static constexpr int NBATCH = 32;
static constexpr int OV = 502;        // valid output size (512 - 11 + 1)
static constexpr int TILE = 16;
static constexpr int NT = 32;         // 32*16 = 512 >= 502, tiles per dim
static constexpr int PATCH = 26;      // 16 + 10 input rows/cols per tile
static constexpr int PSTR = 28;       // padded LDS stride
static constexpr int HROWS = 28;      // 26 rows + 2 zero-weight pad rows (K pad to 28)
static constexpr int WAVES = 4;       // waves (tiles) per workgroup
static constexpr int NTILES = NBATCH * NT * NT;          // 32768
static constexpr int NBLK1 = NTILES / WAVES;             // 8192
static constexpr int NBLK2 = (NBATCH * IMG_H * IMG_W) / 256; // 32768

// normalized 1D gaussian, size 11, sigma 1.5
#define GLIST 0.0010284f, 0.0075988f, 0.0360008f, 0.1093606f, 0.2130055f, \
              0.2660116f, 0.2130055f, 0.1093606f, 0.0360008f, 0.0075988f, 0.0010284f

__device__ __constant__ float cG[11] = { GLIST };

// ---------------------------------------------------------------------------
// Kernel 1: SSIM sums via separable gaussian conv; vertical pass done with
// V_WMMA_F32_16X16X4_F32 chains (banded weight matrix as A, H-filtered rows
// as B). Horizontal pass staged through a small per-wave LDS buffer to keep
// VGPR pressure < 256 (avoids vgpr-msb SALU overhead).
// ---------------------------------------------------------------------------
__global__ __launch_bounds__(128) void ssim_kernel(
    const float* __restrict__ fus, const float* __restrict__ ir,
    const float* __restrict__ vi, float* __restrict__ partials)
{
    __shared__ float sT[WAVES][3][PATCH * PSTR];
    __shared__ float sH[WAVES][HROWS * 16];
    __shared__ float red[128];

    const int lane = threadIdx.x;            // 0..31  (wave32)
    const int wv   = threadIdx.y;            // 0..3
    const int tile = blockIdx.x * WAVES + wv;
    const int b    = tile / (NT * NT);
    const int rt   = (tile / NT) % NT;
    const int ct   = tile % NT;
    const int R0   = rt * TILE;              // first valid output row of tile
    const int C0   = ct * TILE;              // first valid output col of tile
    const long base = (long)b * IMG_H * IMG_W;

    float* tF = sT[wv][0];
    float* tI = sT[wv][1];
    float* tV = sT[wv][2];
    float* hb = sH[wv];

    // load 26x26 input patch (clamped at image edge; clamped data only feeds
    // masked-out outputs)
    for (int idx = lane; idx < PATCH * PATCH; idx += 32) {
        int pr = idx / PATCH;
        int pc = idx - pr * PATCH;
        int gr = R0 + pr; if (gr > IMG_H - 1) gr = IMG_H - 1;
        int gc = C0 + pc; if (gc > IMG_W - 1) gc = IMG_W - 1;
        long off = base + (long)gr * IMG_W + gc;
        tF[pr * PSTR + pc] = fus[off];
        tI[pr * PSTR + pc] = ir[off];
        tV[pr * PSTR + pc] = vi[off];
    }
    __syncthreads();

    const float gl[11] = { GLIST };

    // A fragments: banded vertical weights W[m][k] = g[k-m] for 0<=k-m<=10.
    // f32 16x16x4 A layout: lanes 0-15 hold M=lane,    K = 4t+{0,1};
    //                       lanes 16-31 hold M=lane-16, K = 4t+{2,3}.
    const int m    = lane & 15;
    const int koff = (lane >> 4) << 1;
    v2f Afr[7];
#pragma unroll
    for (int t = 0; t < 7; ++t) {
        int k0 = 4 * t + koff;
        int d0 = k0 - m, d1 = k0 + 1 - m;
        Afr[t].x = (d0 >= 0 && d0 <= 10) ? cG[d0] : 0.0f;
        Afr[t].y = (d1 >= 0 && d1 <= 10) ? cG[d1] : 0.0f;
    }

    const int xcol = lane & 15;   // output column within tile (B-matrix N index)

    // 8 fields: fus, ir, vi, fus^2, ir^2, vi^2, fus*ir, fus*vi
    const float* pA[8] = { tF, tI, tV, tF, tI, tV, tF, tF };
    const float* pB[8] = { nullptr, nullptr, nullptr, tF, tI, tV, tI, tV };

    v8f acc[8];
#pragma unroll
    for (int f = 0; f < 8; ++f) {
        // ---- horizontal gaussian pass: lanes 0..27 each produce one h-row ----
        if (lane < HROWS) {
            int kc = (lane > PATCH - 1) ? (PATCH - 1) : lane; // pad rows dup row 25
            const float* rA = pA[f] + kc * PSTR;
            const float* rB = pB[f] ? (pB[f] + kc * PSTR) : nullptr;
            float p[PATCH];
#pragma unroll
            for (int i = 0; i < PATCH; ++i) {
                float v = rA[i];
                if (pB[f]) v *= rB[i];
                p[i] = v;
            }
#pragma unroll
            for (int x = 0; x < 16; ++x) {
                float s = 0.f;
#pragma unroll
                for (int j = 0; j < 11; ++j) s = fmaf(gl[j], p[x + j], s);
                hb[lane * 16 + x] = s;
            }
        }
        __syncthreads();

        // ---- vertical pass: 7 chained f32 WMMAs (K = 28) ----
        v8f c = { 0.f, 0.f, 0.f, 0.f, 0.f, 0.f, 0.f, 0.f };
#pragma unroll
        for (int t = 0; t < 7; ++t) {
            int k0 = 4 * t + koff;      // f32 4x16 B layout mirrors A lane halves
            v2f bfr;
            bfr.x = hb[k0 * 16 + xcol];
            bfr.y = hb[(k0 + 1) * 16 + xcol];
            c = __builtin_amdgcn_wmma_f32_16x16x4_f32(
                    false, Afr[t], false, bfr, (short)0, c, false, false);
        }
        acc[f] = c;
        __syncthreads();   // hb reused by next field
    }

    // SSIM maps for both pairs from the 8 conv accumulators.
    const float C1c = 0.0001f;   // (0.01)^2
    const float C2c = 0.0009f;   // (0.03)^2
    float lsum = 0.f;
    const int rowBase = R0 + ((lane >> 4) << 3);  // C/D layout: M = e + 8*(lane/16)
    const int col = C0 + xcol;
    if (col < OV) {
#pragma unroll
        for (int e = 0; e < 8; ++e) {
            int row = rowBase + e;
            if (row < OV) {
                float mu1 = acc[0][e], muI = acc[1][e], muV = acc[2][e];
                float Ef2 = acc[3][e], Ei2 = acc[4][e], Ev2 = acc[5][e];
                float Efi = acc[6][e], Efv = acc[7][e];
                float m1s = mu1 * mu1;
                float s1  = Ef2 - m1s;
                {
                    float mIs = muI * muI;
                    float si  = Ei2 - mIs;
                    float sfi = Efi - mu1 * muI;
                    float num = (2.f * mu1 * muI + C1c) * (2.f * sfi + C2c);
                    float den = (m1s + mIs + C1c) * (s1 + si + C2c);
                    lsum += num / den;
                }
                {
                    float mVs = muV * muV;
                    float sv  = Ev2 - mVs;
                    float sfv = Efv - mu1 * muV;
                    float num = (2.f * mu1 * muV + C1c) * (2.f * sfv + C2c);
                    float den = (m1s + mVs + C1c) * (s1 + sv + C2c);
                    lsum += num / den;
                }
            }
        }
    }

    // deterministic block reduction (128 threads)
    const int tid = threadIdx.y * 32 + threadIdx.x;
    red[tid] = lsum;
    __syncthreads();
    for (int s = 64; s > 0; s >>= 1) {
        if (tid < s) red[tid] += red[tid + s];
        __syncthreads();
    }
    if (tid == 0) partials[blockIdx.x] = red[0];
}

// ---------------------------------------------------------------------------
// Kernel 2: L1 term + laplacian gradient term (reflect padding), per-block sums
// ---------------------------------------------------------------------------
__device__ inline int refl(int t) {
    return t < 0 ? -t : (t > IMG_H - 1 ? 2 * (IMG_H - 1) - t : t);
}

__device__ inline float lap3(const float* __restrict__ img, int i, int j) {
    float s = 0.f;
#pragma unroll
    for (int di = -1; di <= 1; ++di) {
        int ri = refl(i + di) * IMG_W;
#pragma unroll
        for (int dj = -1; dj <= 1; ++dj) {
            float v = img[ri + refl(j + dj)];
            s += (di == 0 && dj == 0) ? -8.f * v : v;
        }
    }
    return s * 0.0625f;   // /16 (kornia-normalized laplacian)
}

__global__ __launch_bounds__(256) void l1grad_kernel(
    const float* __restrict__ fus, const float* __restrict__ ir,
    const float* __restrict__ vi, const float* __restrict__ mir,
    const float* __restrict__ mvi,
    float* __restrict__ l1Part, float* __restrict__ grPart)
{
    __shared__ float redA[256];
    __shared__ float redB[256];
    const int id  = blockIdx.x * 256 + threadIdx.x;     // < 32*512*512
    const int b   = id / (IMG_H * IMG_W);
    const int rem = id - b * (IMG_H * IMG_W);
    const int i   = rem / IMG_W;
    const int j   = rem - i * IMG_W;
    const long base = (long)b * IMG_H * IMG_W;

    float l1v = fabsf(fmaxf(mir[id], mvi[id]) - fus[id]);

    float gi = fabsf(lap3(ir + base, i, j));
    float gv = fabsf(lap3(vi + base, i, j));
    float gf = fabsf(lap3(fus + base, i, j));
    float grv = fabsf(fmaxf(gi, gv) - gf);

    const int tid = threadIdx.x;
    redA[tid] = l1v;
    redB[tid] = grv;
    __syncthreads();
    for (int s = 128; s > 0; s >>= 1) {
        if (tid < s) { redA[tid] += redA[tid + s]; redB[tid] += redB[tid + s]; }
        __syncthreads();
    }
    if (tid == 0) {
        l1Part[blockIdx.x] = redA[0];
        grPart[blockIdx.x] = redB[0];
    }
}

// ---------------------------------------------------------------------------
// Kernel 3: deterministic final reduction to the scalar loss
// ---------------------------------------------------------------------------
__global__ __launch_bounds__(256) void finalize_kernel(
    const float* __restrict__ ssimP, const float* __restrict__ l1P,
    const float* __restrict__ grP, float* __restrict__ out)
{
    __shared__ float red[256];
    const int tid = threadIdx.x;
    float S, L, G;

    float s = 0.f;
    for (int i = tid; i < NBLK1; i += 256) s += ssimP[i];
    red[tid] = s; __syncthreads();
    for (int t = 128; t > 0; t >>= 1) { if (tid < t) red[tid] += red[tid + t]; __syncthreads(); }
    S = red[0]; __syncthreads();

    s = 0.f;
    for (int i = tid; i < NBLK2; i += 256) s += l1P[i];
    red[tid] = s; __syncthreads();
    for (int t = 128; t > 0; t >>= 1) { if (tid < t) red[tid] += red[tid + t]; __syncthreads(); }
    L = red[0]; __syncthreads();

    s = 0.f;
    for (int i = tid; i < NBLK2; i += 256) s += grP[i];
    red[tid] = s; __syncthreads();
    for (int t = 128; t > 0; t >>= 1) { if (tid < t) red[tid] += red[tid + t]; __syncthreads(); }
    G = red[0];

    if (tid == 0) {
        const float nSsim = 32.0f * 502.0f * 502.0f;   // 8064128
        const float nPix  = 32.0f * 512.0f * 512.0f;   // 8388608
        float msssim = 2.0f - S / nSsim;               // (1-m1)+(1-m2)
        float l1     = L / nPix;
        float grad   = G / nPix;
        out[0] = msssim + l1 + 10.0f * grad;           // ALPHA=BETA=1, THETA=10
    }
}

// ---------------------------------------------------------------------------
extern "C" void kernel_launch(void* const* d_in, const int* in_sizes, int n_in,
                              void* d_out, int out_size, void* d_ws, size_t ws_size,
                              hipStream_t stream)
{
    const float* fus = (const float*)d_in[0];
    const float* ir  = (const float*)d_in[1];
    const float* vi  = (const float*)d_in[2];
    const float* mir = (const float*)d_in[3];
    const float* mvi = (const float*)d_in[4];

    float* wsf   = (float*)d_ws;
    float* ssimP = wsf;                       // NBLK1 floats
    float* l1P   = wsf + NBLK1;               // NBLK2 floats
    float* grP   = wsf + NBLK1 + NBLK2;       // NBLK2 floats

    dim3 blk1(32, WAVES);
    ssim_kernel<<<NBLK1, blk1, 0, stream>>>(fus, ir, vi, ssimP);
    l1grad_kernel<<<NBLK2, 256, 0, stream>>>(fus, ir, vi, mir, mvi, l1P, grP);
    finalize_kernel<<<1, 256, 0, stream>>>(ssimP, l1P, grP, (float*)d_out);
}